// GCNEdgeNet_50568944943202
// MI455X (gfx1250) — compile-verified
//
#include <hip/hip_runtime.h>
#include <hip/hip_bf16.h>
#include <math.h>

typedef float v2f __attribute__((ext_vector_type(2)));
typedef float v8f __attribute__((ext_vector_type(8)));
typedef unsigned int u32x4 __attribute__((ext_vector_type(4)));
typedef int i32x4 __attribute__((ext_vector_type(4)));
typedef int i32x8 __attribute__((ext_vector_type(8)));

#define LDSA 132   // LDS row stride (floats); 132 mod 64 banks = 4 -> conflict-free b64 frag reads
#define TPB  256

// ---------------------------------------------------------------------------
// TDM: DMA a 2D tile of 4-byte elements into LDS.
//   tile_d0 x tile_d1 elements, memory row stride stride0 (elements),
//   tensor_d0/tensor_d1 bound OOB (OOB reads return zero),
//   pad_code >= 0: insert (pad_code+1) DWORDs after every 128 DWORDs stored.
// Descriptor packing per CDNA5 ISA sections 8.3/8.4.
// ---------------------------------------------------------------------------
__device__ __forceinline__ void tdm_load_2d(uint64_t gaddr, uint32_t lds_byte_off,
                                            uint32_t tile_d0, uint32_t tile_d1,
                                            uint32_t tensor_d0, uint32_t tensor_d1,
                                            uint32_t stride0, int pad_code) {
  u32x4 g0;
  g0[0] = 1u;                                                     // count=1 (user descriptor)
  g0[1] = lds_byte_off;                                           // lds_addr (bytes)
  g0[2] = (uint32_t)gaddr;                                        // global_addr[31:0]
  g0[3] = ((uint32_t)(gaddr >> 32) & 0x01FFFFFFu) | (2u << 30);   // global_addr[56:32] | type=2
  uint32_t d0 = (2u << 16);                                       // data_size = 4 bytes
  if (pad_code >= 0)
    d0 |= (1u << 20) | (6u << 22) | ((uint32_t)pad_code << 25);   // pad every 128 DW
  i32x8 g1;
  g1[0] = (int)d0;
  g1[1] = (int)((tensor_d0 & 0xffffu) << 16);                     // tensor_dim0[15:0]
  g1[2] = (int)(((tensor_d0 >> 16) & 0xffffu) | ((tensor_d1 & 0xffffu) << 16));
  g1[3] = (int)(((tensor_d1 >> 16) & 0xffffu) | ((tile_d0 & 0xffffu) << 16));
  g1[4] = (int)(tile_d1 & 0xffffu);                               // tile_dim1 (tile_dim2=0)
  g1[5] = (int)stride0;                                           // tensor_dim0_stride[31:0]
  g1[6] = 0;
  g1[7] = 0;
  i32x4 z4 = {0, 0, 0, 0};
#if __has_include(<hip/amd_detail/amd_gfx1250_TDM.h>)
  i32x8 z8 = {0, 0, 0, 0, 0, 0, 0, 0};
  __builtin_amdgcn_tensor_load_to_lds(g0, g1, z4, z4, z8, 0);
#else
  __builtin_amdgcn_tensor_load_to_lds(g0, g1, z4, z4, 0);
#endif
}

// ---------------------------------------------------------------------------
// Wave-level 16x128 += 16x128 @ 128x128 with V_WMMA_F32_16X16X4_F32.
// As: [m][k] stride LDSA. Bt: B TRANSPOSED [n][k] stride LDSA.
// Both fragments are contiguous b64 loads per lane -> zero assembly movs.
// ---------------------------------------------------------------------------
__device__ __forceinline__ void wave_gemm_128(const float* As, const float* Bt,
                                              int mbase, int lhalf, int l15,
                                              v8f acc[8]) {
#pragma unroll 4
  for (int kk = 0; kk < 32; ++kk) {
    const int k0 = kk * 4 + lhalf * 2;           // A/B layout: VGPR = K LSB, lane-half = K MSB
    v2f a = *(const v2f*)(As + (mbase + l15) * LDSA + k0);
#pragma unroll
    for (int j = 0; j < 8; ++j) {
      v2f b = *(const v2f*)(Bt + (j * 16 + l15) * LDSA + k0);
      acc[j] = __builtin_amdgcn_wmma_f32_16x16x4_f32(
          false, a, false, b, (short)0, acc[j], false, false);
    }
  }
}

// ---------------------------------------------------------------------------
// Degree / normalization
// ---------------------------------------------------------------------------
__global__ void deg_init_kernel(float* deg, int n) {
  int i = blockIdx.x * blockDim.x + threadIdx.x;
  if (i < n) deg[i] = 1.0f;                       // self loop
}

__global__ void deg_count_kernel(const int* __restrict__ ei, float* deg, int E) {
  int e = blockIdx.x * blockDim.x + threadIdx.x;
  if (e < E) {
    __hip_atomic_fetch_add(&deg[ei[2 * e + 0]], 1.0f, __ATOMIC_RELAXED, __HIP_MEMORY_SCOPE_AGENT);
    __hip_atomic_fetch_add(&deg[ei[2 * e + 1]], 1.0f, __ATOMIC_RELAXED, __HIP_MEMORY_SCOPE_AGENT);
  }
}

__global__ void rsqrt_kernel(float* d, int n) {   // in-place deg -> dinv
  int i = blockIdx.x * blockDim.x + threadIdx.x;
  if (i < n) d[i] = rsqrtf(d[i]);
}

// ---------------------------------------------------------------------------
// H = A @ W (A: R x 128, W: 128 x 128, H: R x 128).
// A tile DMA'd by TDM (pad -> stride 132, OOB rows zeroed) while the waves
// stage W transposed into LDS.  Dynamic LDS: As[128*LDSA] | Wt[128*LDSA]
// ---------------------------------------------------------------------------
__global__ void gemm_node_kernel(const float* __restrict__ A,
                                 const float* __restrict__ W,
                                 float* __restrict__ H, int R) {
  extern __shared__ float lds[];
  float* As = lds;
  float* Wt = lds + 128 * LDSA;
  const int t = threadIdx.x;
  const int row0 = blockIdx.x * 128;

  if (t < 32) {                                   // wave 0 kicks the A-tile DMA first
    uint32_t base = __builtin_amdgcn_groupstaticsize();
    tdm_load_2d((uint64_t)(uintptr_t)(A + (size_t)row0 * 128), base,
                128u, 128u, 128u, (uint32_t)(R - row0), 128u, /*pad=*/3);
  }
  for (int idx = t; idx < 128 * 128; idx += TPB) { // stage W transposed (overlaps TDM)
    int k = idx >> 7, n = idx & 127;
    Wt[n * LDSA + k] = W[idx];
  }
  __builtin_amdgcn_s_wait_tensorcnt(0);
  __syncthreads();

  const int wave = t >> 5, lane = t & 31;
  const int lhalf = lane >> 4, l15 = lane & 15;
  const int mbase = wave * 16;

  v8f acc[8] = {};
  wave_gemm_128(As, Wt, mbase, lhalf, l15, acc);

#pragma unroll
  for (int j = 0; j < 8; ++j)
#pragma unroll
    for (int r = 0; r < 8; ++r) {
      int gr = row0 + mbase + lhalf * 8 + r;      // C/D layout: M = 8*half + vgpr
      if (gr < R) H[(size_t)gr * 128 + j * 16 + l15] = acc[j][r];
    }
}

// F[i,c] = dinv[i]^2 * H[i,c] + b[c]   (self loop + bias)
__global__ void conv_init_kernel(const float* __restrict__ H,
                                 const float* __restrict__ dinv,
                                 const float* __restrict__ b,
                                 float* __restrict__ F, int total) {
  int i = blockIdx.x * blockDim.x + threadIdx.x;
  if (i < total) {
    int node = i >> 7, c = i & 127;
    float di = dinv[node];
    F[i] = di * di * H[i] + b[c];
  }
}

// one wave per edge, both directions: F[v] += n*H[u]; F[u] += n*H[v]
__global__ void scatter_kernel(const float* __restrict__ H,
                               const int* __restrict__ ei,
                               const float* __restrict__ dinv,
                               float* __restrict__ F, int E) {
  int gw = (blockIdx.x * blockDim.x + threadIdx.x) >> 5;
  int lane = threadIdx.x & 31;
  if (gw >= E) return;
  int u = ei[2 * gw + 0], v = ei[2 * gw + 1];
  float nrm = dinv[u] * dinv[v];
  const float* Hu = H + (size_t)u * 128;
  const float* Hv = H + (size_t)v * 128;
  float* Fu = F + (size_t)u * 128;
  float* Fv = F + (size_t)v * 128;
#pragma unroll
  for (int i = 0; i < 4; ++i) {
    int c = lane * 4 + i;
    __hip_atomic_fetch_add(&Fv[c], nrm * Hu[c], __ATOMIC_RELAXED, __HIP_MEMORY_SCOPE_AGENT);
    __hip_atomic_fetch_add(&Fu[c], nrm * Hv[c], __ATOMIC_RELAXED, __HIP_MEMORY_SCOPE_AGENT);
  }
}

__global__ void relu_kernel(float* F, int total) {
  int i = blockIdx.x * blockDim.x + threadIdx.x;
  if (i < total) F[i] = fmaxf(F[i], 0.0f);
}

// ---------------------------------------------------------------------------
// Fused edge MLP: attr = F[src]-F[dst]; relu(attr@Wm1+b1); relu(.@Wm2+b2);
// sigmoid(.@Wm3+b3). 128 edges/block, all intermediates in LDS.
// Edge-index slab DMA'd by TDM while waves stage the transposed weights.
// Dynamic LDS (floats): Sa | G1 | Wt1 | Wt2 (each 128*LDSA) | bias1,bias2,w3 | eis[256]
// ---------------------------------------------------------------------------
#define EOFF_G1   (128 * LDSA)
#define EOFF_WT1  (2 * 128 * LDSA)
#define EOFF_WT2  (3 * 128 * LDSA)
#define EOFF_MISC (4 * 128 * LDSA)
#define EOFF_EIS  (4 * 128 * LDSA + 384)
#define EDGE_LDS_FLOATS (EOFF_EIS + 256)

__global__ void edge_mlp_kernel(const float* __restrict__ F,
                                const int* __restrict__ ei,
                                const float* __restrict__ Wm1, const float* __restrict__ bm1,
                                const float* __restrict__ Wm2, const float* __restrict__ bm2,
                                const float* __restrict__ Wm3, const float* __restrict__ bm3,
                                float* __restrict__ out, int E) {
  extern __shared__ float lds[];
  float* Sa    = lds;                 // attr tile, reused for G2
  float* G1    = lds + EOFF_G1;
  float* Wt1   = lds + EOFF_WT1;
  float* Wt2   = lds + EOFF_WT2;
  float* bias1 = lds + EOFF_MISC;
  float* bias2 = bias1 + 128;
  float* w3    = bias2 + 128;
  int*   eis   = (int*)(lds + EOFF_EIS);          // interleaved (src,dst) pairs

  const int t = threadIdx.x;
  const int e0 = blockIdx.x * 128;

  if (t < 32) {                                   // wave 0: DMA edge-index slab (OOB -> 0)
    uint32_t base = __builtin_amdgcn_groupstaticsize();
    tdm_load_2d((uint64_t)(uintptr_t)(ei + (size_t)2 * e0), base + EOFF_EIS * 4u,
                256u, 1u, 2u * (uint32_t)(E - e0), 1u, 256u, /*pad=*/-1);
  }
  for (int idx = t; idx < 128 * 128; idx += TPB) { // stage both weights transposed (overlaps TDM)
    int k = idx >> 7, n = idx & 127;
    Wt1[n * LDSA + k] = Wm1[idx];
    Wt2[n * LDSA + k] = Wm2[idx];
  }
  if (t < 128) {
    bias1[t] = bm1[t];
    bias2[t] = bm2[t];
    w3[t]    = Wm3[t];
  }
  __builtin_amdgcn_s_wait_tensorcnt(0);
  __syncthreads();                                // eis + weights + biases visible

  for (int idx = t; idx < 128 * 128; idx += TPB) { // gather attr = F[src]-F[dst]
    int m = idx >> 7, k = idx & 127;
    float v = 0.0f;
    if (e0 + m < E)
      v = F[(size_t)eis[2 * m] * 128 + k] - F[(size_t)eis[2 * m + 1] * 128 + k];
    Sa[m * LDSA + k] = v;
  }
  __syncthreads();                                // Sa visible

  const int wave = t >> 5, lane = t & 31;
  const int lhalf = lane >> 4, l15 = lane & 15;
  const int mbase = wave * 16;

  {                                               // layer 1: G1 = relu(Sa@Wm1 + b1)
    v8f acc[8] = {};
    wave_gemm_128(Sa, Wt1, mbase, lhalf, l15, acc);
#pragma unroll
    for (int j = 0; j < 8; ++j)
#pragma unroll
      for (int r = 0; r < 8; ++r) {
        int m = mbase + lhalf * 8 + r, n = j * 16 + l15;
        G1[m * LDSA + n] = fmaxf(acc[j][r] + bias1[n], 0.0f);
      }
  }
  __syncthreads();

  {                                               // layer 2: Sa = relu(G1@Wm2 + b2)
    v8f acc[8] = {};
    wave_gemm_128(G1, Wt2, mbase, lhalf, l15, acc);
#pragma unroll
    for (int j = 0; j < 8; ++j)
#pragma unroll
      for (int r = 0; r < 8; ++r) {
        int m = mbase + lhalf * 8 + r, n = j * 16 + l15;
        Sa[m * LDSA + n] = fmaxf(acc[j][r] + bias2[n], 0.0f);
      }
  }
  __syncthreads();

  if (t < 128 && e0 + t < E) {                    // layer 3: sigmoid(dot(row, w3)+b3)
    float s = 0.0f;
#pragma unroll 8
    for (int k = 0; k < 128; ++k) s += Sa[t * LDSA + k] * w3[k];
    s += bm3[0];
    out[e0 + t] = 1.0f / (1.0f + expf(-s));
  }
}

// ---------------------------------------------------------------------------
extern "C" void kernel_launch(void* const* d_in, const int* in_sizes, int n_in,
                              void* d_out, int out_size, void* d_ws, size_t ws_size,
                              hipStream_t stream) {
  const float* x   = (const float*)d_in[0];
  const int*   ei  = (const int*)  d_in[1];
  const float* W1  = (const float*)d_in[2];
  const float* b1  = (const float*)d_in[3];
  const float* W2  = (const float*)d_in[4];
  const float* b2  = (const float*)d_in[5];
  const float* Wm1 = (const float*)d_in[6];
  const float* bm1 = (const float*)d_in[7];
  const float* Wm2 = (const float*)d_in[8];
  const float* bm2 = (const float*)d_in[9];
  const float* Wm3 = (const float*)d_in[10];
  const float* bm3 = (const float*)d_in[11];
  float* out = (float*)d_out;

  const int N = in_sizes[0] / 128;
  const int E = in_sizes[1] / 2;
  const int total = N * 128;

  // workspace layout (floats): dinv | H | F1 | F2
  size_t degPad = ((size_t)N + 255) & ~(size_t)255;
  size_t matPad = ((size_t)total + 255) & ~(size_t)255;
  if (ws_size < (degPad + 3 * matPad) * sizeof(float)) return;
  float* dinv = (float*)d_ws;
  float* H    = dinv + degPad;
  float* F1   = H + matPad;
  float* F2   = F1 + matPad;

  const int gemmBlocks = (N + 127) / 128;
  const int edgeBlocks = (E + 127) / 128;
  const size_t LDS_GEMM = (size_t)(2 * 128 * LDSA) * sizeof(float);
  const size_t LDS_EDGE = (size_t)EDGE_LDS_FLOATS * sizeof(float);

  // degree + symmetric normalization
  deg_init_kernel<<<(N + TPB - 1) / TPB, TPB, 0, stream>>>(dinv, N);
  deg_count_kernel<<<(E + TPB - 1) / TPB, TPB, 0, stream>>>(ei, dinv, E);
  rsqrt_kernel<<<(N + TPB - 1) / TPB, TPB, 0, stream>>>(dinv, N);

  // GCN layer 1
  gemm_node_kernel<<<gemmBlocks, TPB, LDS_GEMM, stream>>>(x, W1, H, N);
  conv_init_kernel<<<(total + TPB - 1) / TPB, TPB, 0, stream>>>(H, dinv, b1, F1, total);
  scatter_kernel<<<(E * 32 + TPB - 1) / TPB, TPB, 0, stream>>>(H, ei, dinv, F1, E);
  relu_kernel<<<(total + TPB - 1) / TPB, TPB, 0, stream>>>(F1, total);

  // GCN layer 2
  gemm_node_kernel<<<gemmBlocks, TPB, LDS_GEMM, stream>>>(F1, W2, H, N);
  conv_init_kernel<<<(total + TPB - 1) / TPB, TPB, 0, stream>>>(H, dinv, b2, F2, total);
  scatter_kernel<<<(E * 32 + TPB - 1) / TPB, TPB, 0, stream>>>(H, ei, dinv, F2, E);
  relu_kernel<<<(total + TPB - 1) / TPB, TPB, 0, stream>>>(F2, total);

  // fused edge MLP (edge indices via TDM, all intermediates in LDS)
  edge_mlp_kernel<<<edgeBlocks, TPB, LDS_EDGE, stream>>>(
      F2, ei, Wm1, bm1, Wm2, bm2, Wm3, bm3, out, E);
}